// FusionGAT_23021024707098
// MI455X (gfx1250) — compile-verified
//
#include <hip/hip_runtime.h>
#include <math.h>

typedef __attribute__((ext_vector_type(2))) float v2f;
typedef __attribute__((ext_vector_type(8))) float v8f;

#define NEG_SLOPE 0.2f

// float atomic max via signed-max (positives) / unsigned-min (non-positives).
// Destination must be initialized to -inf (0xFF800000). Correct for mixed signs.
__device__ __forceinline__ void atomicMaxF32(float* addr, float val) {
    if (val > 0.0f) atomicMax((int*)addr, __float_as_int(val));
    else            atomicMin((unsigned int*)addr, __float_as_uint(val));
}

__device__ __forceinline__ void edge_nodes(const int* __restrict__ ei, int E, int eid,
                                           int& s, int& d) {
    if (eid < E) { s = ei[eid]; d = ei[E + eid]; }
    else         { s = eid - E; d = s; }          // self loops appended after E
}

__global__ void k_fill(float* __restrict__ p, float v, int n) {
    int t = blockIdx.x * blockDim.x + threadIdx.x;
    if (t < n) p[t] = v;
}

// h1lin[N,64] = x[N,12] @ W1[12,64] using V_WMMA_F32_16X16X4_F32 (K=12 = 3 x K4).
// One wave per 16-node tile; 4 N-tiles of 16 cols. Store guard is wave-uniform:
// full tiles take the unguarded fast path (always, for N % 16 == 0).
__global__ void k_gemm1_wmma(const float* __restrict__ x, const float* __restrict__ W1,
                             float* __restrict__ h1lin, int N, int ntiles) {
    int wave = blockIdx.x * (blockDim.x >> 5) + (threadIdx.x >> 5);
    if (wave >= ntiles) return;                    // uniform per wave
    int lane = threadIdx.x & 31;
    int half = lane >> 4;                          // 0: lanes 0-15, 1: lanes 16-31
    int l16  = lane & 15;
    int row  = wave * 16 + l16;
    int rowc = row < N ? row : N - 1;              // clamp for ragged tail loads
    bool full = (wave * 16 + 16) <= N;             // wave-uniform

    // A 16x4 f32 layout: VGPR0 = K (half*2), VGPR1 = K (half*2+1)
    v2f a[3];
#pragma unroll
    for (int kc = 0; kc < 3; ++kc) {
        int k0 = kc * 4 + half * 2;
        a[kc].x = x[rowc * 12 + k0];
        a[kc].y = x[rowc * 12 + k0 + 1];
    }
#pragma unroll
    for (int nt = 0; nt < 4; ++nt) {
        v8f c = {};
#pragma unroll
        for (int kc = 0; kc < 3; ++kc) {
            // B 4x16 f32: row K striped across lanes; rows split as K= half*2, half*2+1
            int k0 = kc * 4 + half * 2;
            v2f b;
            b.x = W1[k0 * 64 + nt * 16 + l16];
            b.y = W1[(k0 + 1) * 64 + nt * 16 + l16];
            c = __builtin_amdgcn_wmma_f32_16x16x4_f32(false, a[kc], false, b,
                                                      (short)0, c, false, false);
        }
        // C/D: VGPR v -> row v (lanes 0-15) or row v+8 (lanes 16-31), col = l16
        if (full) {
#pragma unroll
            for (int v = 0; v < 8; ++v) {
                int m = wave * 16 + v + half * 8;
                h1lin[m * 64 + nt * 16 + l16] = c[v];
            }
        } else {
#pragma unroll
            for (int v = 0; v < 8; ++v) {
                int m = wave * 16 + v + half * 8;
                if (m < N) h1lin[m * 64 + nt * 16 + l16] = c[v];
            }
        }
    }
}

// per (node, head): attention logits for layer 1
__global__ void k_logits1(const float* __restrict__ h1lin,
                          const float* __restrict__ asrc, const float* __restrict__ adst,
                          float* __restrict__ as1, float* __restrict__ ad1, int n4) {
    int t = blockIdx.x * blockDim.x + threadIdx.x;
    if (t >= n4) return;
    int n = t >> 2, hh = t & 3;
    const float* hp = h1lin + n * 64 + hh * 16;
    const float* ws = asrc + hh * 16;
    const float* wd = adst + hh * 16;
    float s = 0.f, d = 0.f;
#pragma unroll
    for (int c = 0; c < 16; ++c) { float v = hp[c]; s += v * ws[c]; d += v * wd[c]; }
    as1[t] = s; ad1[t] = d;
}

__global__ void k_edge_max1(const int* __restrict__ ei, int E, const float* __restrict__ as1,
                            const float* __restrict__ ad1, float* __restrict__ emax1, int total) {
    int t = blockIdx.x * blockDim.x + threadIdx.x;
    if (t >= total) return;
    int eid = t >> 2, hh = t & 3, s, d;
    edge_nodes(ei, E, eid, s, d);
    float e = as1[s * 4 + hh] + ad1[d * 4 + hh];
    e = e >= 0.f ? e : NEG_SLOPE * e;
    atomicMaxF32(&emax1[d * 4 + hh], e);
}

__global__ void k_edge_sum1(const int* __restrict__ ei, int E, const float* __restrict__ as1,
                            const float* __restrict__ ad1, const float* __restrict__ emax1,
                            float* __restrict__ denom1, int total) {
    int t = blockIdx.x * blockDim.x + threadIdx.x;
    if (t >= total) return;
    int eid = t >> 2, hh = t & 3, s, d;
    edge_nodes(ei, E, eid, s, d);
    float e = as1[s * 4 + hh] + ad1[d * 4 + hh];
    e = e >= 0.f ? e : NEG_SLOPE * e;
    atomicAdd(&denom1[d * 4 + hh], __expf(e - emax1[d * 4 + hh]));
}

__global__ void k_edge_agg1(const int* __restrict__ ei, int E, const float* __restrict__ as1,
                            const float* __restrict__ ad1, const float* __restrict__ emax1,
                            const float* __restrict__ denom1, const float* __restrict__ h1lin,
                            float* __restrict__ agg1, int total) {
    int t = blockIdx.x * blockDim.x + threadIdx.x;
    if (t >= total) return;
    int eid = t >> 2, hh = t & 3, s, d;
    edge_nodes(ei, E, eid, s, d);
    int idx = d * 4 + hh;
    float e = as1[s * 4 + hh] + ad1[idx];
    e = e >= 0.f ? e : NEG_SLOPE * e;
    float alpha = __expf(e - emax1[idx]) / denom1[idx];
    const float* hs = h1lin + s * 64 + hh * 16;
    float* ag = agg1 + d * 64 + hh * 16;
#pragma unroll
    for (int c = 0; c < 16; ++c) atomicAdd(&ag[c], hs[c] * alpha);
}

// per node: h1 = elu(agg1 + b1) (in-place); h2lin = h1 @ W2; layer-2 logits
__global__ void k_node2(float* __restrict__ agg1, const float* __restrict__ b1,
                        const float* __restrict__ W2, const float* __restrict__ a2s,
                        const float* __restrict__ a2d, float* __restrict__ h2lin,
                        float* __restrict__ as2, float* __restrict__ ad2, int N) {
    int n = blockIdx.x * blockDim.x + threadIdx.x;
    if (n >= N) return;
    float o0 = 0.f, o1 = 0.f, o2 = 0.f, o3 = 0.f;
#pragma unroll 4
    for (int k = 0; k < 64; ++k) {
        float v = agg1[n * 64 + k] + b1[k];
        v = v > 0.f ? v : expm1f(v);               // ELU, alpha=1
        agg1[n * 64 + k] = v;                      // becomes h1
        o0 += v * W2[k * 4 + 0];
        o1 += v * W2[k * 4 + 1];
        o2 += v * W2[k * 4 + 2];
        o3 += v * W2[k * 4 + 3];
    }
    h2lin[n * 4 + 0] = o0; h2lin[n * 4 + 1] = o1;
    h2lin[n * 4 + 2] = o2; h2lin[n * 4 + 3] = o3;
    as2[n] = o0 * a2s[0] + o1 * a2s[1] + o2 * a2s[2] + o3 * a2s[3];
    ad2[n] = o0 * a2d[0] + o1 * a2d[1] + o2 * a2d[2] + o3 * a2d[3];
}

__global__ void k_edge_max2(const int* __restrict__ ei, int E, const float* __restrict__ as2,
                            const float* __restrict__ ad2, float* __restrict__ emax2, int total) {
    int eid = blockIdx.x * blockDim.x + threadIdx.x;
    if (eid >= total) return;
    int s, d; edge_nodes(ei, E, eid, s, d);
    float e = as2[s] + ad2[d];
    e = e >= 0.f ? e : NEG_SLOPE * e;
    atomicMaxF32(&emax2[d], e);
}

__global__ void k_edge_sum2(const int* __restrict__ ei, int E, const float* __restrict__ as2,
                            const float* __restrict__ ad2, const float* __restrict__ emax2,
                            float* __restrict__ denom2, float* __restrict__ alpha_out, int total) {
    int eid = blockIdx.x * blockDim.x + threadIdx.x;
    if (eid >= total) return;
    int s, d; edge_nodes(ei, E, eid, s, d);
    float e = as2[s] + ad2[d];
    e = e >= 0.f ? e : NEG_SLOPE * e;
    float p = __expf(e - emax2[d]);
    atomicAdd(&denom2[d], p);
    alpha_out[eid] = p;                            // unnormalized; fixed next pass
}

__global__ void k_edge_agg2(const int* __restrict__ ei, int E, const float* __restrict__ denom2,
                            const float* __restrict__ h2lin, float* __restrict__ agg2,
                            float* __restrict__ alpha_out, int total) {
    int eid = blockIdx.x * blockDim.x + threadIdx.x;
    if (eid >= total) return;
    int s, d; edge_nodes(ei, E, eid, s, d);
    float alpha = alpha_out[eid] / denom2[d];
    alpha_out[eid] = alpha;                        // final alpha2 output
    const float* hs = h2lin + s * 4;
#pragma unroll
    for (int c = 0; c < 4; ++c) atomicAdd(&agg2[d * 4 + c], hs[c] * alpha);
}

__global__ void k_out2(const float* __restrict__ agg2, const float* __restrict__ b2,
                       float* __restrict__ out, int n4) {
    int t = blockIdx.x * blockDim.x + threadIdx.x;
    if (t >= n4) return;
    out[t] = agg2[t] + b2[t & 3];
}

extern "C" void kernel_launch(void* const* d_in, const int* in_sizes, int n_in,
                              void* d_out, int out_size, void* d_ws, size_t ws_size,
                              hipStream_t stream) {
    const float* x     = (const float*)d_in[0];
    const int*   ei    = (const int*)  d_in[1];
    const float* W1    = (const float*)d_in[2];
    const float* asrc1 = (const float*)d_in[3];
    const float* adst1 = (const float*)d_in[4];
    const float* b1    = (const float*)d_in[5];
    const float* W2    = (const float*)d_in[6];
    const float* asrc2 = (const float*)d_in[7];
    const float* adst2 = (const float*)d_in[8];
    const float* b2    = (const float*)d_in[9];

    const int N  = in_sizes[0] / 12;
    const int E  = in_sizes[1] / 2;
    const int Ep = E + N;                          // with self loops

    float* out       = (float*)d_out;              // [N*4] h2
    float* alpha_out = out + (size_t)N * 4;        // [Ep]  alpha2

    float* w      = (float*)d_ws;
    float* h1lin  = w; w += (size_t)N * 64;
    float* as1    = w; w += (size_t)N * 4;
    float* ad1    = w; w += (size_t)N * 4;
    float* emax1  = w; w += (size_t)N * 4;
    float* denom1 = w; w += (size_t)N * 4;
    float* agg1   = w; w += (size_t)N * 64;        // becomes h1 after ELU
    float* h2lin  = w; w += (size_t)N * 4;
    float* as2    = w; w += (size_t)N;
    float* ad2    = w; w += (size_t)N;
    float* emax2  = w; w += (size_t)N;
    float* denom2 = w; w += (size_t)N;
    float* agg2   = w; w += (size_t)N * 4;

    const int B = 256;
    const float NEG_INF = -__builtin_huge_valf();

    // init accumulators (workspace is poisoned; must re-init every call)
    k_fill<<<(N * 64 + B - 1) / B, B, 0, stream>>>(agg1, 0.f, N * 64);
    k_fill<<<(N * 4 + B - 1) / B, B, 0, stream>>>(denom1, 0.f, N * 4);
    k_fill<<<(N * 4 + B - 1) / B, B, 0, stream>>>(emax1, NEG_INF, N * 4);
    k_fill<<<(N * 4 + B - 1) / B, B, 0, stream>>>(agg2, 0.f, N * 4);
    k_fill<<<(N + B - 1) / B, B, 0, stream>>>(denom2, 0.f, N);
    k_fill<<<(N + B - 1) / B, B, 0, stream>>>(emax2, NEG_INF, N);

    // layer-1 linear via f32 WMMA
    int ntiles = (N + 15) / 16;
    k_gemm1_wmma<<<(ntiles + 3) / 4, 128, 0, stream>>>(x, W1, h1lin, N, ntiles);

    // layer-1 attention
    k_logits1<<<(N * 4 + B - 1) / B, B, 0, stream>>>(h1lin, asrc1, adst1, as1, ad1, N * 4);
    k_edge_max1<<<(Ep * 4 + B - 1) / B, B, 0, stream>>>(ei, E, as1, ad1, emax1, Ep * 4);
    k_edge_sum1<<<(Ep * 4 + B - 1) / B, B, 0, stream>>>(ei, E, as1, ad1, emax1, denom1, Ep * 4);
    k_edge_agg1<<<(Ep * 4 + B - 1) / B, B, 0, stream>>>(ei, E, as1, ad1, emax1, denom1,
                                                        h1lin, agg1, Ep * 4);

    // ELU + layer-2 linear + logits
    k_node2<<<(N + B - 1) / B, B, 0, stream>>>(agg1, b1, W2, asrc2, adst2, h2lin, as2, ad2, N);

    // layer-2 attention (alpha2 written straight into output region)
    k_edge_max2<<<(Ep + B - 1) / B, B, 0, stream>>>(ei, E, as2, ad2, emax2, Ep);
    k_edge_sum2<<<(Ep + B - 1) / B, B, 0, stream>>>(ei, E, as2, ad2, emax2, denom2, alpha_out, Ep);
    k_edge_agg2<<<(Ep + B - 1) / B, B, 0, stream>>>(ei, E, denom2, h2lin, agg2, alpha_out, Ep);

    // bias add -> h2 output
    k_out2<<<(N * 4 + B - 1) / B, B, 0, stream>>>(agg2, b2, out, N * 4);
}